// GNN_49100066128394
// MI455X (gfx1250) — compile-verified
//
#include <hip/hip_runtime.h>

typedef float v2f __attribute__((ext_vector_type(2)));
typedef float v8f __attribute__((ext_vector_type(8)));

// Native fire-and-forget f32 atomic add (global_atomic_add_f32, no return).
__device__ __forceinline__ void atomadd(float* p, float v) {
    unsafeAtomicAdd(p, v);
}

// ---------------------------------------------------------------------------
// Zero-fill (graph-capture-safe replacement for hipMemsetAsync)
// ---------------------------------------------------------------------------
__global__ void gcn_zero_f32(float* __restrict__ p, long long n) {
    long long i = (long long)blockIdx.x * blockDim.x + threadIdx.x;
    if (i < n) p[i] = 0.0f;
}

// ---------------------------------------------------------------------------
// Degree accumulation: deg[dst] += 1 per edge (self-loops folded in later)
// ---------------------------------------------------------------------------
__global__ void gcn_degree(const long long* __restrict__ ei, float* __restrict__ deg,
                           long long E) {
    long long e = (long long)blockIdx.x * blockDim.x + threadIdx.x;
    if (e < E) {
        int d = (int)ei[E + e];
        atomadd(deg + d, 1.0f);
    }
}

// dinv[i] = rsqrt(deg[i] + 1)   (+1 = self-loop, so deg > 0 always)
__global__ void gcn_dinv(float* __restrict__ deg, int n) {
    int i = blockIdx.x * blockDim.x + threadIdx.x;
    if (i < n) deg[i] = __frsqrt_rn(deg[i] + 1.0f);
}

// ---------------------------------------------------------------------------
// Dense GEMM H = X @ W via V_WMMA_F32_16X16X4_F32.
// Block = 256 threads = 8 waves; each wave owns a 16(M) x OUT_C tile,
// block covers 128 rows. W staged in LDS *transposed* with padded pitch so
// each B fragment is one aligned ds_load_b64 (K-contiguous), conflict-free.
// A frag (16x4 f32): lane l%16 = row M; lanes<16 carry K={k,k+1}, lanes>=16
// carry K={k+2,k+3} in vgprs {0,1}. B frag mirrored (lane = col N).
// C/D: vgpr r -> M = r + 8*(lane>=16), N = lane%16.
// ---------------------------------------------------------------------------
template <int IN_C, int OUT_C>
__global__ void gcn_gemm_wmma(const float* __restrict__ X, const float* __restrict__ W,
                              float* __restrict__ H, int nrows) {
    constexpr int P = IN_C + 2;               // padded pitch (even -> b64 aligned)
    __shared__ float sWt[OUT_C * P];          // sWt[col * P + k] = W[k][col]
    const int tid = threadIdx.x;
    for (int i = tid; i < IN_C * OUT_C; i += 256) {
        const int k = i / OUT_C;
        const int c = i - k * OUT_C;          // coalesced global read of W[k][c]
        sWt[c * P + k] = W[i];
    }
    __syncthreads();

    const int wave = tid >> 5;
    const int lane = tid & 31;
    const int half = lane >> 4;   // 0: K pair {0,1}; 1: K pair {2,3}
    const int l16  = lane & 15;

    const int rowBase = blockIdx.x * 128 + wave * 16;
    const int arow    = rowBase + l16;
    const int arowc   = arow < nrows ? arow : (nrows - 1);   // clamp, keep EXEC all-1

    constexpr int NT = OUT_C / 16;
    v8f acc[NT] = {};

    const float* __restrict__ xrow = X + (long long)arowc * IN_C;

#pragma unroll 4
    for (int k0 = 0; k0 < IN_C; k0 += 4) {
        const int kb = k0 + half * 2;
        const v2f a = *(const v2f*)(xrow + kb);
#pragma unroll
        for (int n = 0; n < NT; ++n) {
            const v2f b = *(const v2f*)(sWt + (n * 16 + l16) * P + kb);  // ds_load_b64
            acc[n] = __builtin_amdgcn_wmma_f32_16x16x4_f32(
                false, a, false, b, (short)0, acc[n], false, false);
        }
    }

    float* __restrict__ hbase = H + (long long)(rowBase + half * 8) * OUT_C + l16;
    if (rowBase + 16 <= nrows) {
        // fast path: full tile in range -> unguarded coalesced stores
#pragma unroll
        for (int n = 0; n < NT; ++n)
#pragma unroll
            for (int r = 0; r < 8; ++r)
                hbase[(long long)r * OUT_C + n * 16] = acc[n][r];
    } else {
        // boundary tile only (last wave of last block)
#pragma unroll
        for (int n = 0; n < NT; ++n)
#pragma unroll
            for (int r = 0; r < 8; ++r) {
                const int row = rowBase + half * 8 + r;
                if (row < nrows) H[(long long)row * OUT_C + n * 16 + l16] = acc[n][r];
            }
    }
}

// ---------------------------------------------------------------------------
// Edge scatter, 64 channels: one wave covers 2 edges; 16 lanes per edge,
// float4 (b128) per lane.  agg[dst][c] += h[src][c] * dinv[src] * dinv[dst]
// ---------------------------------------------------------------------------
__global__ void gcn_scatter64(const float* __restrict__ H, const float* __restrict__ dinv,
                              const long long* __restrict__ ei, float* __restrict__ agg,
                              long long E) {
    long long t = (long long)blockIdx.x * blockDim.x + threadIdx.x;
    long long e = (t >> 5) * 2 + ((t >> 4) & 1);   // 2 edges per wave
    int l16 = (int)(t & 15);
    if (e >= E) return;
    int s = (int)ei[e];
    int d = (int)ei[E + e];
    float nrm = dinv[s] * dinv[d];
    float4 m = *(const float4*)(H + (long long)s * 64 + l16 * 4);
    float* a = agg + (long long)d * 64 + l16 * 4;
    atomadd(a + 0, m.x * nrm);
    atomadd(a + 1, m.y * nrm);
    atomadd(a + 2, m.z * nrm);
    atomadd(a + 3, m.w * nrm);
}

// Edge scatter, 32 channels: one wave covers 4 edges; 8 lanes per edge,
// float4 (b128) per lane.
__global__ void gcn_scatter32(const float* __restrict__ H, const float* __restrict__ dinv,
                              const long long* __restrict__ ei, float* __restrict__ agg,
                              long long E) {
    long long t = (long long)blockIdx.x * blockDim.x + threadIdx.x;
    long long e = (t >> 5) * 4 + ((t >> 3) & 3);   // 4 edges per wave
    int l8 = (int)(t & 7);
    if (e >= E) return;
    int s = (int)ei[e];
    int d = (int)ei[E + e];
    float nrm = dinv[s] * dinv[d];
    float4 m = *(const float4*)(H + (long long)s * 32 + l8 * 4);
    float* a = agg + (long long)d * 32 + l8 * 4;
    atomadd(a + 0, m.x * nrm);
    atomadd(a + 1, m.y * nrm);
    atomadd(a + 2, m.z * nrm);
    atomadd(a + 3, m.w * nrm);
}

// ---------------------------------------------------------------------------
// Epilogues: fold self-loop (h[i]*dinv[i]^2) + bias (+ relu for layer 1).
// Layer 1 writes y1 in place over agg1.
// ---------------------------------------------------------------------------
__global__ void gcn_finish_relu(const float* __restrict__ H, const float* __restrict__ dinv,
                                const float* __restrict__ bias, float* __restrict__ agg,
                                int n, int C) {
    long long t = (long long)blockIdx.x * blockDim.x + threadIdx.x;
    if (t >= (long long)n * C) return;
    int node = (int)(t / C);
    int c    = (int)(t - (long long)node * C);
    float di = dinv[node];
    float v  = agg[t] + H[t] * di * di + bias[c];
    agg[t] = v > 0.0f ? v : 0.0f;
}

__global__ void gcn_finish(const float* __restrict__ H, const float* __restrict__ dinv,
                           const float* __restrict__ bias, float* __restrict__ out,
                           int n, int C) {
    long long t = (long long)blockIdx.x * blockDim.x + threadIdx.x;
    if (t >= (long long)n * C) return;
    int node = (int)(t / C);
    int c    = (int)(t - (long long)node * C);
    float di = dinv[node];
    out[t] = out[t] + H[t] * di * di + bias[c];
}

// ---------------------------------------------------------------------------
// Launch
// ---------------------------------------------------------------------------
extern "C" void kernel_launch(void* const* d_in, const int* in_sizes, int n_in,
                              void* d_out, int out_size, void* d_ws, size_t ws_size,
                              hipStream_t stream) {
    const float*      x  = (const float*)d_in[0];
    const long long*  ei = (const long long*)d_in[1];
    const float*      W1 = (const float*)d_in[2];
    const float*      b1 = (const float*)d_in[3];
    const float*      W2 = (const float*)d_in[4];
    const float*      b2 = (const float*)d_in[5];

    const int       HID   = in_sizes[3];             // 64
    const int       OUTC  = in_sizes[5];             // 32
    const int       INC   = in_sizes[2] / HID;       // 128
    const int       N     = in_sizes[0] / INC;       // 100000
    const long long E     = (long long)in_sizes[1] / 2;

    // workspace layout (256B aligned)
    auto align256 = [](size_t v) { return (v + 255) & ~(size_t)255; };
    char* ws = (char*)d_ws;
    size_t off = 0;
    float* deg  = (float*)(ws + off); off = align256(off + (size_t)N * 4);          // -> dinv in place
    float* h1   = (float*)(ws + off); off = align256(off + (size_t)N * HID * 4);
    float* agg1 = (float*)(ws + off); off = align256(off + (size_t)N * HID * 4);    // -> y1 in place
    float* h2   = (float*)(ws + off); off = align256(off + (size_t)N * OUTC * 4);
    (void)ws_size;

    float* out = (float*)d_out;

    const int B = 256;
    // ---- zero deg, agg1, out (agg2 accumulates directly in d_out) ----
    gcn_zero_f32<<<(N + B - 1) / B, B, 0, stream>>>(deg, N);
    gcn_zero_f32<<<((long long)N * HID + B - 1) / B, B, 0, stream>>>(agg1, (long long)N * HID);
    gcn_zero_f32<<<((long long)N * OUTC + B - 1) / B, B, 0, stream>>>(out, (long long)N * OUTC);

    // ---- degree + dinv ----
    gcn_degree<<<(unsigned)((E + B - 1) / B), B, 0, stream>>>(ei, deg, E);
    gcn_dinv<<<(N + B - 1) / B, B, 0, stream>>>(deg, N);

    // ---- layer 1: h1 = x @ W1 (WMMA) ----
    gcn_gemm_wmma<128, 64><<<(N + 127) / 128, 256, 0, stream>>>(x, W1, h1, N);

    // ---- layer 1: edge scatter (2 edges / wave, b128 gathers) ----
    {
        long long threads = E * 16;
        gcn_scatter64<<<(unsigned)((threads + B - 1) / B), B, 0, stream>>>(h1, deg, ei, agg1, E);
    }
    // ---- layer 1 epilogue: self-loop + bias + relu -> y1 (in agg1) ----
    gcn_finish_relu<<<(unsigned)(((long long)N * HID + B - 1) / B), B, 0, stream>>>(
        h1, deg, b1, agg1, N, HID);

    // ---- layer 2: h2 = y1 @ W2 (WMMA) ----
    gcn_gemm_wmma<64, 32><<<(N + 127) / 128, 256, 0, stream>>>(agg1, W2, h2, N);

    // ---- layer 2: edge scatter into d_out (4 edges / wave, b128 gathers) ----
    {
        long long threads = E * 8;
        gcn_scatter32<<<(unsigned)((threads + B - 1) / B), B, 0, stream>>>(h2, deg, ei, out, E);
    }
    // ---- layer 2 epilogue: self-loop + bias ----
    gcn_finish<<<(unsigned)(((long long)N * OUTC + B - 1) / B), B, 0, stream>>>(
        h2, deg, b2, out, N, OUTC);
}